// InferenceLayer_10539849744797
// MI455X (gfx1250) — compile-verified
//
#include <hip/hip_runtime.h>
#include <cstdint>
#include <cstddef>

typedef __attribute__((ext_vector_type(2))) float v2f;
typedef __attribute__((ext_vector_type(8))) float v8f;

#define Bn 8
#define Ln 128
#define Dn 768
#define NPOS (Bn * Ln * Ln)     /* 131072 */
#define PER_BATCH (Ln * Ln)     /* 16384  */
#define WPB 8                   /* waves per block */
#define BLOCK1 (WPB * 32)       /* 256 threads */
#define TILES (NPOS / 16)       /* 8192 */
#define GRID1 (TILES / WPB)     /* 1024 blocks */

// ---------------------------------------------------------------------------
// Kernel 1: single streaming pass over table (402 MB, NT loads).
// Each wave computes 16 positions' logits_S/logits_E via V_WMMA_F32_16X16X4_F32,
// then BCE partials + sigmoid preds.
// ---------------------------------------------------------------------------
__global__ __launch_bounds__(BLOCK1) void logits_bce_kernel(
    const float* __restrict__ table,
    const int* __restrict__ labels_S,
    const int* __restrict__ labels_E,
    const float* __restrict__ W_S, const float* __restrict__ b_S,
    const float* __restrict__ W_E, const float* __restrict__ b_E,
    float* __restrict__ pred_S, float* __restrict__ pred_E,
    float* __restrict__ partial_S, float* __restrict__ partial_E)
{
    __shared__ float lw[2][Dn];          // W_S, W_E staged in LDS
    __shared__ float lsS[WPB][16];
    __shared__ float lsE[WPB][16];
    __shared__ float red[BLOCK1];

    const int tid = threadIdx.x;
    for (int i = tid; i < Dn; i += BLOCK1) { lw[0][i] = W_S[i]; lw[1][i] = W_E[i]; }
    __syncthreads();

    const int wave = tid >> 5;
    const int lane = tid & 31;
    const int half = lane >> 4;          // K-group: 0 -> K{0,1}, 1 -> K{2,3}
    const int col  = lane & 15;          // A row (M) / B column (N)

    const int tile = blockIdx.x * WPB + wave;
    const size_t row = (size_t)tile * 16 + (size_t)col;
    const float* arow = table + row * Dn + 2 * half;
    // B column 0 = W_S, column 1 = W_E; other columns read W_S (results unused)
    const float* wrow = (col == 1) ? lw[1] : lw[0];

    v8f c = {};                          // 16x16 f32 accumulator (8 VGPRs)
#pragma unroll 8
    for (int k = 0; k < Dn; k += 4) {
        v2f a = __builtin_nontemporal_load((const v2f*)(arow + k));   // single-use stream
        v2f b = *(const v2f*)(wrow + k + 2 * half);                   // LDS broadcast
        c = __builtin_amdgcn_wmma_f32_16x16x4_f32(
                /*neg_a=*/false, a, /*neg_b=*/false, b,
                /*c_mod=*/(short)0, c, /*reuse_a=*/false, /*reuse_b=*/false);
    }

    // D layout: VGPR r, lane-half h -> M = r + 8*h, N = col. S in N==0, E in N==1.
    if (col == 0) {
#pragma unroll
        for (int r = 0; r < 8; ++r) lsS[wave][half * 8 + r] = c[r];
    } else if (col == 1) {
#pragma unroll
        for (int r = 0; r < 8; ++r) lsE[wave][half * 8 + r] = c[r];
    }
    __syncthreads();

    float lossS = 0.0f, lossE = 0.0f;
    if (lane < 16) {
        const size_t p = (size_t)tile * 16 + (size_t)lane;
        const float lgS = lsS[wave][lane] + b_S[0];
        const float lgE = lsE[wave][lane] + b_E[0];
        const int ls = labels_S[p];
        const int le = labels_E[p];
        const float w  = (ls >= 0) ? 1.0f : 0.0f;
        // stable logaddexp(0,x) - x*t  ==  max(x,0)+log1p(exp(-|x|)) - x*t
        const float bS = fmaxf(lgS, 0.0f) + log1pf(expf(-fabsf(lgS))) - lgS * (float)ls;
        const float bE = fmaxf(lgE, 0.0f) + log1pf(expf(-fabsf(lgE))) - lgE * (float)le;
        lossS = w * bS;
        lossE = w * bE;
        pred_S[p] = w / (1.0f + expf(-lgS));
        pred_E[p] = w / (1.0f + expf(-lgE));
    }

    // deterministic block tree-reduction of loss partials
    red[tid] = lossS;
    __syncthreads();
    for (int s = BLOCK1 / 2; s > 0; s >>= 1) {
        if (tid < s) red[tid] += red[tid + s];
        __syncthreads();
    }
    if (tid == 0) partial_S[blockIdx.x] = red[0];
    __syncthreads();
    red[tid] = lossE;
    __syncthreads();
    for (int s = BLOCK1 / 2; s > 0; s >>= 1) {
        if (tid < s) red[tid] += red[tid + s];
        __syncthreads();
    }
    if (tid == 0) partial_E[blockIdx.x] = red[0];
}

// ---------------------------------------------------------------------------
// Kernel 2: fold 1024 block partials -> loss scalars (deterministic order)
// ---------------------------------------------------------------------------
__global__ __launch_bounds__(256) void loss_finalize(
    const float* __restrict__ pS, const float* __restrict__ pE,
    float* __restrict__ out)
{
    __shared__ float red[256];
    const int tid = threadIdx.x;
    float a = 0.0f, b = 0.0f;
    for (int i = tid; i < GRID1; i += 256) { a += pS[i]; b += pE[i]; }
    red[tid] = a;
    __syncthreads();
    for (int s = 128; s > 0; s >>= 1) {
        if (tid < s) red[tid] += red[tid + s];
        __syncthreads();
    }
    if (tid == 0) out[0] = red[0] / (float)NPOS;
    __syncthreads();
    red[tid] = b;
    __syncthreads();
    for (int s = 128; s > 0; s >>= 1) {
        if (tid < s) red[tid] += red[tid + s];
        __syncthreads();
    }
    if (tid == 0) out[1] = red[0] / (float)NPOS;
}

// ---------------------------------------------------------------------------
// Kernel 3: exact k-th largest per (tensor, batch) via 4-pass radix select
// on float bit patterns (all preds >= 0, so uint order == float order).
// ---------------------------------------------------------------------------
__global__ __launch_bounds__(256) void topk_select(
    const float* __restrict__ predS, const float* __restrict__ predE,
    const int* __restrict__ amask, float* __restrict__ thresh)
{
    __shared__ int hist[256];
    __shared__ int sh_k;
    __shared__ unsigned sh_prefix;

    const int t = blockIdx.x;            // 0..15
    const int tensor = t >> 3, b = t & 7;
    const float* data = (tensor ? predE : predS) + (size_t)b * PER_BATCH;
    const int tid = threadIdx.x;

    if (tid == 0) {
        int ml = 0;
        for (int i = 0; i < Ln; ++i) ml += amask[b * Ln + i];
        ml -= 2;                                   // mask_length
        int len = (int)((float)ml * 0.3f);         // trunc, like .long()
        if (len < 5) len = 5;
        const int cap = ml * ml;
        if (len > cap) len = cap;
        if (len < 1) len = 1;
        sh_k = len;
        sh_prefix = 0u;
    }
    __syncthreads();

    unsigned pmask = 0u;
    for (int shift = 24; shift >= 0; shift -= 8) {
        hist[tid] = 0;
        __syncthreads();
        const unsigned prefix = sh_prefix;
        for (int i = tid; i < PER_BATCH; i += 256) {
            const unsigned u = __float_as_uint(data[i]);
            if ((u & pmask) == prefix)
                atomicAdd(&hist[(u >> shift) & 255], 1);
        }
        __syncthreads();
        if (tid == 0) {
            int k = sh_k, sel = 0;
            for (int d = 255; d >= 0; --d) {
                const int cnt = hist[d];
                if (cnt >= k) { sel = d; break; }
                k -= cnt;
            }
            sh_k = k;
            sh_prefix = prefix | ((unsigned)sel << shift);
        }
        __syncthreads();
        pmask |= (0xFFu << shift);
    }
    if (tid == 0) thresh[t] = __uint_as_float(sh_prefix);
}

// ---------------------------------------------------------------------------
// Kernel 4: predict = pred >= kth_value  (1.0f / 0.0f)
// ---------------------------------------------------------------------------
__global__ __launch_bounds__(256) void predicate_kernel(
    const float* __restrict__ predS, const float* __restrict__ predE,
    const float* __restrict__ thresh, float* __restrict__ out)
{
    const int idx = blockIdx.x * blockDim.x + threadIdx.x;
    if (idx >= 2 * NPOS) return;
    const int tensor = idx / NPOS;
    const int r = idx - tensor * NPOS;
    const int b = r / PER_BATCH;
    const float v  = (tensor ? predE : predS)[r];
    const float th = thresh[tensor * 8 + b];
    out[2 + idx] = (v >= th) ? 1.0f : 0.0f;
}

// ---------------------------------------------------------------------------
extern "C" void kernel_launch(void* const* d_in, const int* in_sizes, int n_in,
                              void* d_out, int out_size, void* d_ws, size_t ws_size,
                              hipStream_t stream)
{
    (void)in_sizes; (void)n_in; (void)out_size; (void)ws_size;
    const float* table = (const float*)d_in[0];
    const int*   amask = (const int*)d_in[1];
    const int*   labS  = (const int*)d_in[2];
    const int*   labE  = (const int*)d_in[3];
    const float* W_S   = (const float*)d_in[4];
    const float* b_S   = (const float*)d_in[5];
    const float* W_E   = (const float*)d_in[6];
    const float* b_E   = (const float*)d_in[7];
    float* out = (float*)d_out;

    float* ws      = (float*)d_ws;
    float* pred_S  = ws;                    // NPOS
    float* pred_E  = ws + NPOS;             // NPOS
    float* part_S  = ws + 2 * NPOS;         // GRID1
    float* part_E  = part_S + GRID1;        // GRID1
    float* thresh  = part_E + GRID1;        // 16

    hipLaunchKernelGGL(logits_bce_kernel, dim3(GRID1), dim3(BLOCK1), 0, stream,
                       table, labS, labE, W_S, b_S, W_E, b_E,
                       pred_S, pred_E, part_S, part_E);
    hipLaunchKernelGGL(loss_finalize, dim3(1), dim3(256), 0, stream,
                       part_S, part_E, out);
    hipLaunchKernelGGL(topk_select, dim3(16), dim3(256), 0, stream,
                       pred_S, pred_E, amask, thresh);
    const int n2 = 2 * NPOS;
    hipLaunchKernelGGL(predicate_kernel, dim3((n2 + 255) / 256), dim3(256), 0, stream,
                       pred_S, pred_E, thresh, out);
}